// Gat_11295763988535
// MI455X (gfx1250) — compile-verified
//
#include <hip/hip_runtime.h>
#include <hip/hip_bf16.h>

// ---------------------------------------------------------------------------
// GATv2 layer: N=50000 nodes, E=800000 edges, F_IN=128, C=300
// Inputs (d_in order): x, edge_index(i64), Wl, bl, Wr, br, att, bias, lw, lb
// Output: sigmoid(relu(GATv2(x)) @ lw + lb)  -> [N] floats
// ---------------------------------------------------------------------------

#define GAT_N     50000
#define GAT_E     800000
#define GAT_FIN   128
#define GAT_C     300
#define GAT_CP    304          // C padded to 16 for WMMA B loads
#define NEG_SLOPE 0.2f

typedef __attribute__((ext_vector_type(2))) float v2f;
typedef __attribute__((ext_vector_type(8))) float v8f;

// Order-preserving float <-> uint encoding for atomic segment-max.
__device__ __forceinline__ unsigned enc_f32(float f) {
    unsigned u = __float_as_uint(f);
    return (u & 0x80000000u) ? ~u : (u | 0x80000000u);
}
__device__ __forceinline__ float dec_f32(unsigned u) {
    return (u & 0x80000000u) ? __uint_as_float(u & 0x7FFFFFFFu)
                             : __uint_as_float(~u);
}

// ---------------------------------------------------------------------------
// 0a) Zero the accumulators each call (atomics require clean state).
// ---------------------------------------------------------------------------
__global__ void init_ws_kernel(float* __restrict__ outacc,
                               float* __restrict__ denom,
                               unsigned* __restrict__ menc) {
    size_t i = (size_t)blockIdx.x * blockDim.x + threadIdx.x;
    size_t tot = (size_t)GAT_N * GAT_C;
    if (i < tot) outacc[i] = 0.0f;
    if (i < (size_t)GAT_N) { denom[i] = 0.0f; menc[i] = 0u; }
}

// ---------------------------------------------------------------------------
// 0b) Pad W [128,300] -> Wp [128,304] (zero-filled tail columns) so the GEMM
//     B-fragment loads need no bounds check (no EXEC divergence in k-loop).
// ---------------------------------------------------------------------------
__global__ void pad_w_kernel(const float* __restrict__ Wl,
                             const float* __restrict__ Wr,
                             float* __restrict__ Wlp,
                             float* __restrict__ Wrp) {
    int i = blockIdx.x * blockDim.x + threadIdx.x;      // 0 .. 128*304-1
    if (i >= GAT_FIN * GAT_CP) return;
    int k = i / GAT_CP;
    int c = i - k * GAT_CP;
    float vl = 0.0f, vr = 0.0f;
    if (c < GAT_C) {
        vl = Wl[k * GAT_C + c];
        vr = Wr[k * GAT_C + c];
    }
    Wlp[i] = vl;
    Wrp[i] = vr;
}

// ---------------------------------------------------------------------------
// 1) Dense GEMM  Y[M,C] = X[M,K] @ W[K,C] + b   via V_WMMA_F32_16X16X4_F32.
//    One wave per 16x16 output tile; K fully unrolled in steps of 4.
//    fp32 A 16x4 layout (ISA 7.12.2): lanes 0-15 hold K=k,k+1; lanes 16-31
//    hold K=k+2,k+3. B 4x16: per VGPR, lane half selects K row, N = lane&15.
//    C/D: VGPR r -> M = r (+8 for lane-high half).
//    W is the padded [128,304] copy -> unconditional B loads, constant-offset
//    address streams (compiler emits immediate-offset global_load clauses).
// ---------------------------------------------------------------------------
__global__ void gemm_wmma_f32(const float* __restrict__ X,
                              const float* __restrict__ Wp,   // [128, 304]
                              const float* __restrict__ bias, // [300]
                              float* __restrict__ Y) {        // [N, 300]
    const int wave = blockIdx.x * (blockDim.x >> 5) + (threadIdx.x >> 5);
    const int lane = threadIdx.x & 31;
    const int tilesN = GAT_CP >> 4;                    // 19
    const int tm = wave / tilesN;
    const int tn = wave - tm * tilesN;
    if (tm * 16 >= GAT_N) return;                      // uniform per wave

    const int half = lane >> 4;                        // 0 | 1
    const int l15  = lane & 15;
    const int row  = tm * 16 + l15;                    // N % 16 == 0
    const int col  = tn * 16 + l15;                    // < 304 always

    // A: one b64 load per step (8B aligned: k%4==0, +2*half even)
    const float* __restrict__ pa = X + (size_t)row * GAT_FIN + 2 * half;
    // B: two b32 streams, stride 4*304 floats per step
    const float* __restrict__ pb0 = Wp + (size_t)half * GAT_CP + col;
    const float* __restrict__ pb1 = Wp + (size_t)(2 + half) * GAT_CP + col;

    v8f acc = {};
#pragma unroll
    for (int kk = 0; kk < GAT_FIN / 4; ++kk) {
        v2f a = *(const v2f*)(pa + 4 * kk);
        v2f b;
        b.x = pb0[4 * GAT_CP * kk];
        b.y = pb1[4 * GAT_CP * kk];
        acc = __builtin_amdgcn_wmma_f32_16x16x4_f32(
            /*neg_a=*/false, a, /*neg_b=*/false, b,
            /*c_mod=*/(short)0, acc, /*reuse_a=*/false, /*reuse_b=*/false);
    }

    if (col < GAT_C) {
        const float bb = bias[col];
        float* __restrict__ py = Y + (size_t)(tm * 16 + half * 8) * GAT_C + col;
#pragma unroll
        for (int r = 0; r < 8; ++r)
            py[(size_t)r * GAT_C] = acc[r] + bb;
    }
}

// ---------------------------------------------------------------------------
// 2) Per-edge attention logit: e = att . leakyrelu(xl[src] + xr[dst]).
//    One wave per edge; lane-strided dot, shfl reduction, atomic segment-max.
// ---------------------------------------------------------------------------
__global__ void edge_logit_kernel(const long long* __restrict__ ei,
                                  const float* __restrict__ xl,
                                  const float* __restrict__ xr,
                                  const float* __restrict__ att,
                                  float* __restrict__ ebuf,
                                  unsigned* __restrict__ menc) {
    const int e = blockIdx.x * (blockDim.x >> 5) + (threadIdx.x >> 5);
    const int lane = threadIdx.x & 31;
    if (e >= GAT_E + GAT_N) return;

    int src, dst;
    if (e < GAT_E) { src = (int)ei[e]; dst = (int)ei[GAT_E + e]; }
    else           { src = dst = e - GAT_E; }           // self loop

    const float* __restrict__ pl = xl + (size_t)src * GAT_C;
    const float* __restrict__ pr = xr + (size_t)dst * GAT_C;
    float s = 0.0f;
#pragma unroll
    for (int it = 0; it < 10; ++it) {                   // ceil(300/32) = 10
        int c = lane + 32 * it;
        if (c < GAT_C) {
            float v = pl[c] + pr[c];
            v = (v > 0.0f) ? v : NEG_SLOPE * v;         // leaky_relu
            s = fmaf(v, att[c], s);
        }
    }
#pragma unroll
    for (int off = 16; off > 0; off >>= 1) s += __shfl_down(s, off, 32);
    if (lane == 0) {
        ebuf[e] = s;
        atomicMax(menc + dst, enc_f32(s));
    }
}

// ---------------------------------------------------------------------------
// 3) exp(e - max) and denominator accumulation.
// ---------------------------------------------------------------------------
__global__ void edge_exp_kernel(const long long* __restrict__ ei,
                                float* __restrict__ ebuf,
                                const unsigned* __restrict__ menc,
                                float* __restrict__ denom) {
    const int e = blockIdx.x * blockDim.x + threadIdx.x;
    if (e >= GAT_E + GAT_N) return;
    const int dst = (e < GAT_E) ? (int)ei[GAT_E + e] : (e - GAT_E);
    const float ex = __expf(ebuf[e] - dec_f32(menc[dst]));
    ebuf[e] = ex;
    atomicAdd(denom + dst, ex);
}

// ---------------------------------------------------------------------------
// 4) Weighted scatter: out[dst] += ex * xl[src]. One block (320 thr) per edge;
//    all 300 channels covered in a single pass of L2-resident atomics.
// ---------------------------------------------------------------------------
__global__ void edge_scatter_kernel(const long long* __restrict__ ei,
                                    const float* __restrict__ xl,
                                    const float* __restrict__ ebuf,
                                    float* __restrict__ outacc) {
    const int e = blockIdx.x;
    int src, dst;
    if (e < GAT_E) { src = (int)ei[e]; dst = (int)ei[GAT_E + e]; }
    else           { src = dst = e - GAT_E; }
    const float w = ebuf[e];
    const int c = threadIdx.x;
    if (c < GAT_C)
        atomicAdd(outacc + (size_t)dst * GAT_C + c,
                  w * xl[(size_t)src * GAT_C + c]);
}

// ---------------------------------------------------------------------------
// 5) Per-node finalize: relu(acc/denom + bias) . lw, sigmoid. One wave/node.
// ---------------------------------------------------------------------------
__global__ void finalize_kernel(const float* __restrict__ outacc,
                                const float* __restrict__ denom,
                                const float* __restrict__ bias,
                                const float* __restrict__ lw,
                                const float* __restrict__ lb,
                                float* __restrict__ out) {
    const int n = blockIdx.x * (blockDim.x >> 5) + (threadIdx.x >> 5);
    const int lane = threadIdx.x & 31;
    if (n >= GAT_N) return;
    const float inv = 1.0f / denom[n];                 // self-loop => denom > 0
    const float* __restrict__ po = outacc + (size_t)n * GAT_C;
    float s = 0.0f;
#pragma unroll
    for (int it = 0; it < 10; ++it) {
        int c = lane + 32 * it;
        if (c < GAT_C) {
            float h = fmaf(po[c], inv, bias[c]);
            h = (h > 0.0f) ? h : 0.0f;                 // relu
            s = fmaf(h, lw[c], s);
        }
    }
#pragma unroll
    for (int off = 16; off > 0; off >>= 1) s += __shfl_down(s, off, 32);
    if (lane == 0)
        out[n] = 1.0f / (1.0f + __expf(-(s + lb[0])));
}

// ---------------------------------------------------------------------------
// Launch
// ---------------------------------------------------------------------------
extern "C" void kernel_launch(void* const* d_in, const int* in_sizes, int n_in,
                              void* d_out, int out_size, void* d_ws, size_t ws_size,
                              hipStream_t stream) {
    const float*     x    = (const float*)d_in[0];
    const long long* ei   = (const long long*)d_in[1];   // int64 [2, E]
    const float*     Wl   = (const float*)d_in[2];
    const float*     bl   = (const float*)d_in[3];
    const float*     Wr   = (const float*)d_in[4];
    const float*     br   = (const float*)d_in[5];
    const float*     att  = (const float*)d_in[6];
    const float*     bias = (const float*)d_in[7];
    const float*     lw   = (const float*)d_in[8];
    const float*     lb   = (const float*)d_in[9];
    float*           out  = (float*)d_out;

    // Workspace (floats): xl | xr | outacc | ebuf | denom | menc | Wlp | Wrp
    const size_t NC = (size_t)GAT_N * GAT_C;            // 15,000,000
    float*    xl     = (float*)d_ws;
    float*    xr     = xl + NC;
    float*    outacc = xr + NC;
    float*    ebuf   = outacc + NC;
    float*    denom  = ebuf + (GAT_E + GAT_N);
    unsigned* menc   = (unsigned*)(denom + GAT_N);
    float*    Wlp    = (float*)(menc + GAT_N);
    float*    Wrp    = Wlp + (size_t)GAT_FIN * GAT_CP;

    const int totE = GAT_E + GAT_N;                     // 850,000

    // 0a) zero accumulators
    {
        int blk = 256;
        int grd = (int)((NC + blk - 1) / blk);
        init_ws_kernel<<<grd, blk, 0, stream>>>(outacc, denom, menc);
    }
    // 0b) pad weights to [128,304]
    {
        int tot = GAT_FIN * GAT_CP;
        pad_w_kernel<<<(tot + 255) / 256, 256, 0, stream>>>(Wl, Wr, Wlp, Wrp);
    }
    // 1) two WMMA GEMMs: xl = x@Wl + bl, xr = x@Wr + br
    {
        const int waves = (GAT_N / 16) * (GAT_CP / 16); // 3125 * 19 = 59,375
        const int blk = 256;                            // 8 waves / block
        const int grd = (waves + 7) / 8;
        gemm_wmma_f32<<<grd, blk, 0, stream>>>(x, Wlp, bl, xl);
        gemm_wmma_f32<<<grd, blk, 0, stream>>>(x, Wrp, br, xr);
    }
    // 2) edge logits + segment max
    {
        const int grd = (totE + 7) / 8;                 // 8 waves / block
        edge_logit_kernel<<<grd, 256, 0, stream>>>(ei, xl, xr, att, ebuf, menc);
    }
    // 3) exp + denom
    {
        const int grd = (totE + 255) / 256;
        edge_exp_kernel<<<grd, 256, 0, stream>>>(ei, ebuf, menc, denom);
    }
    // 4) weighted scatter (one 320-thread block per edge)
    {
        edge_scatter_kernel<<<totE, 320, 0, stream>>>(ei, xl, ebuf, outacc);
    }
    // 5) finalize per node
    {
        const int grd = (GAT_N + 7) / 8;
        finalize_kernel<<<grd, 256, 0, stream>>>(outacc, denom, bias, lw, lb, out);
    }
}